// GraphConvolution_3822520893865
// MI455X (gfx1250) — compile-verified
//
#include <hip/hip_runtime.h>

typedef __attribute__((ext_vector_type(2))) float v2f;
typedef __attribute__((ext_vector_type(8))) float v8f;

#define BATCH 8
#define IN_F  128
#define OUT_F 128
#define BOUT  (BATCH * OUT_F)   // 1024 columns of support/out
#define MT    5                 // 16-row tiles per wave (80 rows): 50000 = 625*80
#define MROWS (MT * 16)

// ---------------------------------------------------------------------------
// Kernel 1: batched dense GEMM via V_WMMA_F32_16X16X4_F32.
// support[n, b*128 + p] = sum_k x[n,b,k] * w[b,k,p]
// Block = 256 threads = 8 waves; wave w handles p-tile w (p0 = 16*w).
// blockIdx.x encodes (m_supertile, batch).
// K loop is FULLY unrolled so the 32 preloaded B fragments (64 VGPRs) keep
// compile-time-constant indices and never spill to scratch.
// ---------------------------------------------------------------------------
__global__ __launch_bounds__(256) void gcn_gemm_wmma(
    const float* __restrict__ x, const float* __restrict__ w,
    float* __restrict__ sup, int n_nodes)
{
    const int wave = threadIdx.x >> 5;       // 0..7 -> p-tile
    const int lane = threadIdx.x & 31;
    const int half = lane >> 4;              // 0: K={0,1}, 1: K={2,3}
    const int lm   = lane & 15;              // M (for A) / N (for B) index

    const int n_super  = n_nodes / MROWS;    // 625
    const int m_super  = blockIdx.x % n_super;
    const int b        = blockIdx.x / n_super;
    const int p0       = wave * 16;
    const int m0       = m_super * MROWS;

    const float* __restrict__ wb = w + (size_t)b * IN_F * OUT_F;

    // Preload B fragments for the whole K dimension (32 k-steps of 4).
    // Lane layout per ISA: VGPR0 holds rows K=(4k + 2*half), VGPR1 rows K+1.
    v2f bf[IN_F / 4];
#pragma unroll
    for (int ks = 0; ks < IN_F / 4; ++ks) {
        const int kb = ks * 4 + half * 2;
        bf[ks].x = wb[(size_t)kb * OUT_F + p0 + lm];
        bf[ks].y = wb[(size_t)(kb + 1) * OUT_F + p0 + lm];
    }

    v8f acc[MT];
#pragma unroll
    for (int t = 0; t < MT; ++t) acc[t] = (v8f){0, 0, 0, 0, 0, 0, 0, 0};

    // Per-tile A row base pointers (advance by 4 floats per k-step).
    const float* __restrict__ xr[MT];
#pragma unroll
    for (int t = 0; t < MT; ++t) {
        const int m = m0 + t * 16 + lm;
        xr[t] = x + ((size_t)m * BATCH + b) * IN_F + half * 2;
    }

#pragma unroll
    for (int ks = 0; ks < IN_F / 4; ++ks) {
#pragma unroll
        for (int t = 0; t < MT; ++t) {
            v2f af;                 // contiguous pair -> global_load_b64
            af.x = xr[t][ks * 4 + 0];
            af.y = xr[t][ks * 4 + 1];
            // 8 args: (neg_a, A, neg_b, B, c_mod, C, reuse_a, reuse_b)
            acc[t] = __builtin_amdgcn_wmma_f32_16x16x4_f32(
                false, af, false, bf[ks], (short)0, acc[t], false, false);
        }
    }

    // D layout: VGPR r, lanes 0-15 -> M=r, lanes 16-31 -> M=r+8; N = lane%16.
#pragma unroll
    for (int t = 0; t < MT; ++t) {
#pragma unroll
        for (int r = 0; r < 8; ++r) {
            const int m = m0 + t * 16 + half * 8 + r;
            sup[(size_t)m * BOUT + b * OUT_F + p0 + lm] = acc[t][r];
        }
    }
}

// ---------------------------------------------------------------------------
// Kernel 2: COO SpMM scatter. One 256-thread block per edge; each thread
// moves one float4 (block covers the 1024-float row), scaled by edge_val,
// accumulated with global_atomic_add_f32 (L2-resident RMW).
// ---------------------------------------------------------------------------
__global__ __launch_bounds__(256) void gcn_spmm(
    const int* __restrict__ erow, const int* __restrict__ ecol,
    const float* __restrict__ eval_, const float* __restrict__ sup,
    float* __restrict__ out, int n_edges)
{
    const int e = blockIdx.x;
    if (e >= n_edges) return;
    const int   r = erow[e];     // uniform -> scalar loads
    const int   c = ecol[e];
    const float v = eval_[e];

    const float4 g =
        ((const float4*)(sup + (size_t)c * BOUT))[threadIdx.x];
    float* o = out + (size_t)r * BOUT + (size_t)threadIdx.x * 4;
    atomicAdd(o + 0, g.x * v);
    atomicAdd(o + 1, g.y * v);
    atomicAdd(o + 2, g.z * v);
    atomicAdd(o + 3, g.w * v);
}

// ---------------------------------------------------------------------------
// Kernel 0: zero the output (atomic accumulation target).
// ---------------------------------------------------------------------------
__global__ __launch_bounds__(256) void gcn_zero(float4* __restrict__ out,
                                                size_t n4)
{
    size_t i = (size_t)blockIdx.x * blockDim.x + threadIdx.x;
    if (i < n4) out[i] = make_float4(0.f, 0.f, 0.f, 0.f);
}

extern "C" void kernel_launch(void* const* d_in, const int* in_sizes, int n_in,
                              void* d_out, int out_size, void* d_ws,
                              size_t ws_size, hipStream_t stream)
{
    const float* x     = (const float*)d_in[0];
    const float* w     = (const float*)d_in[1];
    const int*   erow  = (const int*)d_in[2];
    const int*   ecol  = (const int*)d_in[3];
    const float* eval_ = (const float*)d_in[4];
    float*       out   = (float*)d_out;
    float*       sup   = (float*)d_ws;   // [n_nodes, 1024] fp32 ~ 205 MB

    const int n_nodes = in_sizes[0] / (BATCH * IN_F);   // 50000
    const int n_edges = in_sizes[2];                     // 800000

    // 0) zero the accumulation target
    const size_t n4 = (size_t)out_size / 4;
    const int zblocks = (int)((n4 + 255) / 256);
    gcn_zero<<<zblocks, 256, 0, stream>>>((float4*)out, n4);

    // 1) WMMA GEMM -> support in workspace
    const int n_super = n_nodes / MROWS;                 // 625
    gcn_gemm_wmma<<<dim3(n_super * BATCH), 256, 0, stream>>>(x, w, sup,
                                                             n_nodes);

    // 2) edge-parallel atomic SpMM
    gcn_spmm<<<dim3(n_edges), 256, 0, stream>>>(erow, ecol, eval_, sup, out,
                                                n_edges);
}